// GraphConvolution_1589137899808
// MI455X (gfx1250) — compile-verified
//
#include <hip/hip_runtime.h>

typedef __attribute__((ext_vector_type(2))) float v2f;
typedef __attribute__((ext_vector_type(8))) float v8f;

#define N_NODES 100000
#define N_EDGES 1200000
#define D_IN    256
#define D_OUT   64
#define NPAIR   (D_IN / 2)   // 128 K-pair rows

// -------------------------------------------------------------------------
// Kernel 1: support = x @ W via V_WMMA_F32_16X16X4_F32 (full FP32 tensor op).
// Block = 256 threads = 8 waves; each wave owns a 16-row x 64-col tile.
// W staged in LDS as float2 K-pairs {W[2p][n], W[2p+1][n]} so every B
// fragment is one aligned ds_load_b64 straight into an even VGPR pair.
// Column index XOR-swizzled by (p&1)*16 so the two half-wave lane groups
// (pair rows p and p+1) hit disjoint LDS bank halves.
// -------------------------------------------------------------------------
__global__ __launch_bounds__(256) void gcn_gemm_wmma(const float* __restrict__ x,
                                                     const float* __restrict__ w,
                                                     float* __restrict__ support) {
    __shared__ float2 wlds[NPAIR * D_OUT];   // 64 KB

    const int tid = threadIdx.x;
    // stage W: pair consecutive K rows per column, apply bank swizzle
    for (int i = tid; i < NPAIR * D_OUT; i += 256) {
        const int p = i >> 6;          // K-pair row: covers K = 2p, 2p+1
        const int n = i & 63;          // output column
        float2 v;
        v.x = w[(2 * p)     * D_OUT + n];
        v.y = w[(2 * p + 1) * D_OUT + n];
        wlds[(p << 6) | (n ^ ((p & 1) << 4))] = v;
    }
    __syncthreads();

    const int wave = tid >> 5;
    const int lane = tid & 31;
    const int r0   = (blockIdx.x * 8 + wave) * 16;   // 16 rows per wave
    if (r0 >= N_NODES) return;                       // wave-uniform exit

    const int m     = lane & 15;          // A row within tile / B,D column base
    const int khalf = (lane >> 4) * 2;    // 0 (lanes 0-15) or 2 (lanes 16-31)

    v8f acc0 = {}, acc1 = {}, acc2 = {}, acc3 = {};

    const float* xrow = x + (size_t)(r0 + m) * D_IN + khalf;

    for (int k = 0; k < D_IN; k += 4) {
        // A fragment: contiguous pair -> single global_load_b64
        v2f a = *(const v2f*)(xrow + k);

        // B fragments: pair row for this lane group, swizzled columns
        const int     pb = (k >> 1) + (lane >> 4);      // k/2 or k/2 + 1
        const int     sw = (pb & 1) << 4;               // XOR bank swizzle
        const float2* wk = wlds + (pb << 6) + m;        // + column base n=m
        v2f b0 = *(const v2f*)(wk + (0  ^ sw));
        v2f b1 = *(const v2f*)(wk + (16 ^ sw));
        v2f b2 = *(const v2f*)(wk + (32 ^ sw));
        v2f b3 = *(const v2f*)(wk + (48 ^ sw));

        acc0 = __builtin_amdgcn_wmma_f32_16x16x4_f32(false, a, false, b0,
                                                     (short)0, acc0, false, false);
        acc1 = __builtin_amdgcn_wmma_f32_16x16x4_f32(false, a, false, b1,
                                                     (short)0, acc1, false, false);
        acc2 = __builtin_amdgcn_wmma_f32_16x16x4_f32(false, a, false, b2,
                                                     (short)0, acc2, false, false);
        acc3 = __builtin_amdgcn_wmma_f32_16x16x4_f32(false, a, false, b3,
                                                     (short)0, acc3, false, false);
    }

    // D layout: VGPR v -> row r0+v (lanes 0-15) / r0+v+8 (lanes 16-31)
    const int n     = lane & 15;
    const int rbase = r0 + (lane >> 4) * 8;
    #pragma unroll
    for (int v = 0; v < 8; ++v) {
        float* dst = support + (size_t)(rbase + v) * D_OUT + n;
        dst[0]  = acc0[v];
        dst[16] = acc1[v];
        dst[32] = acc2[v];
        dst[48] = acc3[v];
    }
}

// -------------------------------------------------------------------------
// Kernel 2: zero-init output (harness poisons d_out), float4-vectorized
// -------------------------------------------------------------------------
__global__ void gcn_zero4(float4* __restrict__ out, int n4) {
    int i = blockIdx.x * blockDim.x + threadIdx.x;
    if (i < n4) out[i] = make_float4(0.f, 0.f, 0.f, 0.f);
}

// -------------------------------------------------------------------------
// Kernel 3: out[row[e]] += val[e] * support[col[e]]
// 64 consecutive threads (2 waves) per edge. Edge index forced into an SGPR
// via readfirstlane -> row/col/val become scalar (SMEM) loads, one per wave
// instead of 32 redundant vector loads. Gather + atomics fully coalesced;
// support (25.6 MB) is resident in MI455X's 192 MB L2.
// -------------------------------------------------------------------------
__global__ __launch_bounds__(256) void gcn_scatter(const long long* __restrict__ erow,
                                                   const long long* __restrict__ ecol,
                                                   const float* __restrict__ eval_,
                                                   const float* __restrict__ support,
                                                   float* __restrict__ out) {
    const int tid  = threadIdx.x;
    const int lane = tid & 31;
    const int wid  = tid >> 5;                       // 8 waves = 4 edges per block
    const unsigned e = __builtin_amdgcn_readfirstlane(
        blockIdx.x * 4u + (unsigned)(wid >> 1));     // wave-uniform edge id
    const int c = ((wid & 1) << 5) | lane;           // column 0..63

    const long long r  = erow[e];
    const long long cc = ecol[e];
    const float     v  = eval_[e];

    const float msg = v * support[cc * D_OUT + c];
    atomicAdd(out + r * D_OUT + c, msg);
}

// -------------------------------------------------------------------------
// Kernel 4: ReLU in place, float4-vectorized
// -------------------------------------------------------------------------
__global__ void gcn_relu4(float4* __restrict__ out, int n4) {
    int i = blockIdx.x * blockDim.x + threadIdx.x;
    if (i < n4) {
        float4 v = out[i];
        v.x = fmaxf(v.x, 0.f); v.y = fmaxf(v.y, 0.f);
        v.z = fmaxf(v.z, 0.f); v.w = fmaxf(v.w, 0.f);
        out[i] = v;
    }
}

// -------------------------------------------------------------------------
extern "C" void kernel_launch(void* const* d_in, const int* in_sizes, int n_in,
                              void* d_out, int out_size, void* d_ws, size_t ws_size,
                              hipStream_t stream) {
    const float*     x        = (const float*)d_in[0];
    const long long* edge_row = (const long long*)d_in[1];
    const long long* edge_col = (const long long*)d_in[2];
    const float*     edge_val = (const float*)d_in[3];
    const float*     weight   = (const float*)d_in[4];
    float*           out      = (float*)d_out;
    float*           support  = (float*)d_ws;        // 25.6 MB scratch

    const int total = N_NODES * D_OUT;
    const int n4    = total / 4;                     // divisible: 6.4M / 4

    // zero the accumulation target
    gcn_zero4<<<(n4 + 255) / 256, 256, 0, stream>>>((float4*)out, n4);

    // dense GEMM on the WMMA pipe: 128 rows per block
    const int gemm_blocks = (N_NODES + 127) / 128;
    gcn_gemm_wmma<<<gemm_blocks, 256, 0, stream>>>(x, weight, support);

    // sparse scatter-add: 4 edges per 256-thread block (N_EDGES % 4 == 0)
    gcn_scatter<<<N_EDGES / 4, 256, 0, stream>>>(
        edge_row, edge_col, edge_val, support, out);

    // final activation
    gcn_relu4<<<(n4 + 255) / 256, 256, 0, stream>>>((float4*)out, n4);
}